// LatentFlowMambaModel_54820962566297
// MI455X (gfx1250) — compile-verified
//
#include <hip/hip_runtime.h>

// ---------------------------------------------------------------------------
// LatentFlowMamba model on gfx1250.
//   * All convs (strided / circular / transposed) -> one implicit-GEMM kernel
//     on V_WMMA_F32_16X16X4_F32 (fp32-exact), tap-major K loop, dual-M tiles.
//   * SSM state update (the bandwidth hog: ~1.5GB streamed over the run)
//     -> fused roll+update+readout+max streaming kernel with look-ahead
//       global_prefetch of the next shift's state rows.
// ---------------------------------------------------------------------------

#define F_CIRC     1
#define F_TRANS    2
#define F_BN       4
#define F_ADD      8
#define F_LRELU    16
#define F_SOFTPLUS 32
#define F_SIGMOID  64

typedef __attribute__((ext_vector_type(2))) float v2f;
typedef __attribute__((ext_vector_type(8))) float v8f;

__device__ __forceinline__ float conv_epilogue(
    float v, int m, const float* bias, const float* gamma, const float* beta,
    const float* res, long rb, long oix, float ext, int flags)
{
    if (bias) v += bias[m];
    if (flags & F_BN) v = v * (gamma[m] * 0.9999950000374996f) + beta[m];
    if (flags & F_ADD) v += res[rb + oix];
    if (flags & F_LRELU) v = (v >= 0.f) ? v : 0.1f * v;
    if (flags & F_SOFTPLUS) { v += ext; v = (v > 20.f) ? v : log1pf(expf(v)); }
    if (flags & F_SIGMOID) v = 1.f / (1.f + expf(-v));
    return v;
}

// ---------------------------------------------------------------------------
// Implicit-GEMM conv:  out[m, p] = sum_k W[m,k] * X[k, p]  (+ epilogue)
//   k = tap*Cin + ci ; tap-major loop hoists all coordinate math, the hot
//   ci-loop is stride-load + WMMA only. When Cout%32==0 a wave computes a
//   32(M) x 16(N) tile: one B fragment feeds two WMMAs.
// fp32 WMMA layouts (ISA 7.12.2): A lane=m,(vgpr,half)->k ; B/C/D lane=n.
// ---------------------------------------------------------------------------
__global__ void conv_wmma_kernel(
    const float* __restrict__ x, long xbs,
    const float* __restrict__ wgt,
    const float* __restrict__ bias,
    const float* __restrict__ gamma, const float* __restrict__ beta,
    const float* __restrict__ res, long rbs,
    const float* __restrict__ extra_ptr,
    float* __restrict__ out, long obs,
    int Bn, int Cin, int Cout,
    int Hin, int Win, int Hout, int Wout,
    int KH, int KW, int stride, int pad, int flags)
{
    const int wave = threadIdx.x >> 5;
    const int lane = threadIdx.x & 31;
    const bool trans = (flags & F_TRANS) != 0;
    const bool circ  = (flags & F_CIRC)  != 0;
    const bool dualm = ((Cout & 31) == 0) && (Cin >= 4);
    const int Mtiles = dualm ? (Cout >> 5) : ((Cout + 15) >> 4);
    const int Ntiles = (Bn * Hout * Wout) >> 4;
    const int tile = blockIdx.x * (blockDim.x >> 5) + wave;
    if (tile >= Mtiles * Ntiles) return;

    const int mt  = tile % Mtiles;
    const int nt  = tile / Mtiles;
    const int n0  = nt << 4;
    const int HWo = Hout * Wout;
    const int b   = n0 / HWo;
    const int pr  = n0 - b * HWo;
    const int oh  = pr / Wout;            // Wout multiple of 16 -> single row
    const int ow0 = pr - oh * Wout;
    const int half = lane >> 4;
    const int l16  = lane & 15;
    const int ow   = ow0 + l16;           // this lane's output pixel (N index)
    const int mstep = dualm ? 32 : 16;
    int m_a = mt * mstep + l16;           // this lane's A-matrix row (M index)
    if (m_a >= Cout) m_a = Cout - 1;      // clamp: surplus D rows never stored
    const int taps = KH * KW;
    const int HW   = Hin * Win;
    const float* xb = x + (long)b * xbs;

    v8f acc  = {0.f,0.f,0.f,0.f,0.f,0.f,0.f,0.f};
    v8f acc2 = {0.f,0.f,0.f,0.f,0.f,0.f,0.f,0.f};

    if (Cin >= 4) {
        // per-ci strides: weight OIHW -> taps ; transposed IOHW -> Cout*taps
        const long astride = trans ? (long)Cout * taps : (long)taps;
        const long abase0  = trans ? (long)m_a * taps  : (long)m_a * Cin * taps;
        const long am16    = trans ? (long)16 * taps   : (long)16 * Cin * taps;

        for (int tap = 0; tap < taps; ++tap) {
            const int kh = tap / KW;
            const int kw = tap - kh * KW;
            bool inb;
            long bbase;
            if (!trans) {
                int ih = oh * stride - pad + kh;
                int iw = ow * stride - pad + kw;
                if (circ) {                       // circular pad (pow2 dims)
                    ih = (ih + Hin) & (Hin - 1);
                    iw = (iw + Win) & (Win - 1);
                    inb = true;
                } else {
                    inb = (ih >= 0 && ih < Hin && iw >= 0 && iw < Win);
                }
                bbase = (long)ih * Win + iw;
            } else {                              // ConvTranspose2d k=4,s=2,p=1
                int nh = oh + pad - kh;
                int nw = ow + pad - kw;
                int ih = nh >> 1, iw = nw >> 1;
                inb = (nh >= 0 && nw >= 0 && ((nh | nw) & 1) == 0 &&
                       ih < Hin && iw < Win);
                bbase = (long)ih * Win + iw;
            }
            const float* ap = wgt + abase0 + tap;
            const float* bp = xb + bbase;

#pragma unroll 4
            for (int ci = 0; ci < Cin; ci += 4) {
                const int c0 = ci + (half << 1);  // fp32 A/B: vgprs -> c0,c0+1
                v2f av, bv;
                av[0] = ap[(long)c0 * astride];
                av[1] = ap[(long)(c0 + 1) * astride];
                bv[0] = inb ? bp[(long)c0 * HW] : 0.f;
                bv[1] = inb ? bp[(long)(c0 + 1) * HW] : 0.f;
                acc = __builtin_amdgcn_wmma_f32_16x16x4_f32(
                          false, av, false, bv, (short)0, acc, false, false);
                if (dualm) {
                    v2f av2;
                    av2[0] = ap[am16 + (long)c0 * astride];
                    av2[1] = ap[am16 + (long)(c0 + 1) * astride];
                    acc2 = __builtin_amdgcn_wmma_f32_16x16x4_f32(
                               false, av2, false, bv, (short)0, acc2, false, false);
                }
            }
        }
    } else {
        // Cin == 1 (first encoder conv only): k == tap, zero-pad, not transposed
        for (int kk = 0; kk < taps; kk += 4) {
            const int k0 = kk + (half << 1);
            v2f av, bv;
#pragma unroll
            for (int c2 = 0; c2 < 2; ++c2) {
                const int tap = k0 + c2;
                const int kh = tap / KW;
                const int kw = tap - kh * KW;
                av[c2] = wgt[(long)m_a * taps + tap];
                int ih = oh * stride - pad + kh;
                int iw = ow * stride - pad + kw;
                bv[c2] = (ih >= 0 && ih < Hin && iw >= 0 && iw < Win)
                             ? xb[(long)ih * Win + iw] : 0.f;
            }
            acc = __builtin_amdgcn_wmma_f32_16x16x4_f32(
                      false, av, false, bv, (short)0, acc, false, false);
        }
    }

    // Epilogue: bias -> eval-BN -> residual add -> lrelu / softplus(+dt) / sigmoid
    const float ext = extra_ptr ? extra_ptr[0] : 0.f;
    const long rb = (long)b * rbs;
#pragma unroll
    for (int i = 0; i < 8; ++i) {
        const int m = mt * mstep + i + (half << 3);   // D layout: vgpr i, half
        if (m >= Cout) continue;
        const long oix = ((long)m * Hout + oh) * Wout + ow;
        out[(long)b * obs + oix] =
            conv_epilogue(acc[i], m, bias, gamma, beta, res, rb, oix, ext, flags);
    }
    if (dualm) {
#pragma unroll
        for (int i = 0; i < 8; ++i) {
            const int m = mt * mstep + 16 + i + (half << 3);
            const long oix = ((long)m * Hout + oh) * Wout + ow;
            out[(long)b * obs + oix] =
                conv_epilogue(acc2[i], m, bias, gamma, beta, res, rb, oix, ext, flags);
        }
    }
}

// ---------------------------------------------------------------------------
// A_bar / E precompute: hoists exp() out of the 49-way shift loop.
// ---------------------------------------------------------------------------
__global__ void mamba_prep(const float* __restrict__ delta,
                           const float* __restrict__ Bv,
                           const float* __restrict__ u,
                           const float* __restrict__ logA,
                           float* __restrict__ Abar, float* __restrict__ E)
{
    int i = blockIdx.x * blockDim.x + threadIdx.x;    // 2*64*8*256 = 262144
    if (i >= 262144) return;
    int p = i & 255;
    int s = (i >> 8) & 7;
    int c = (i >> 11) & 63;
    int b = i >> 17;
    float A  = -expf(logA[c * 8 + s]);
    float d  = delta[(b * 64 + c) * 256 + p];
    float Ab = expf(d * A);
    float e  = (Ab - 1.f) / A * Bv[(b * 8 + s) * 256 + p]
                             * u[(b * 64 + c) * 256 + p];
    Abar[i] = Ab;
    E[i]    = e;
}

// ---------------------------------------------------------------------------
// Fused SSM state stream (bandwidth-bound hot loop):
//   s_next[v] = A_bar * roll_v(s_prev[v]) + E       (write)
//   y         = max_v ( sum_s s_next[v,s]*Cv[s] ) + u*D
// One block per (b,c); one thread per pixel; 8 states in registers; 49 shifts.
// Next shift's state rows are prefetched one iteration ahead
// (global_prefetch_b8) to hide HBM/L2 latency behind the FMA chain.
// ---------------------------------------------------------------------------
__global__ void mamba_state(const float* __restrict__ sprev,
                            float* __restrict__ snext,
                            const float* __restrict__ Abar,
                            const float* __restrict__ E,
                            const float* __restrict__ Cv,
                            const float* __restrict__ u,
                            const float* __restrict__ Dp,
                            float* __restrict__ y, int first)
{
    const int b = blockIdx.x >> 6;
    const int c = blockIdx.x & 63;
    const int p = threadIdx.x;           // 256 = 16x16 pixels
    const int h = p >> 4, w = p & 15;

    float ab[8], e[8], cv[8];
    const int base = ((b * 64 + c) * 8) * 256 + p;
#pragma unroll
    for (int s = 0; s < 8; ++s) {
        ab[s] = Abar[base + s * 256];
        e[s]  = E[base + s * 256];
        cv[s] = Cv[(b * 8 + s) * 256 + p];
    }
    const float du = u[(b * 64 + c) * 256 + p] * Dp[c];

    float ymax = -3.4e38f;
    for (int v = 0; v < 49; ++v) {
        // V_LIST = [(x,y) for x in -3..3 for y in -3..3]; roll by (vy,vx)
        const int vx = v / 7 - 3;
        const int vy = v % 7 - 3;
        const int hp = (h - vy) & 15;
        const int wp = (w - vx) & 15;
        const int p2 = (hp << 4) | wp;
        const long sb = ((long)((b * 49 + v) * 64 + c)) << 11;   // *8*256

        if (!first && v + 1 < 49) {      // look-ahead prefetch of next shift
            const long sbn = sb + ((long)64 << 11);              // v+1 block
#pragma unroll
            for (int s = 0; s < 8; ++s)
                __builtin_prefetch(&sprev[sbn + s * 256 + p2], 0, 3);
        }

        float acc = 0.f;
#pragma unroll
        for (int s = 0; s < 8; ++s) {
            float sp = first ? 0.f : sprev[sb + s * 256 + p2];
            float sn = fmaf(ab[s], sp, e[s]);
            snext[sb + s * 256 + p] = sn;
            acc = fmaf(sn, cv[s], acc);
        }
        ymax = fmaxf(ymax, acc);
    }
    y[(b * 64 + c) * 256 + p] = ymax + du;
}

// ---------------------------------------------------------------------------
// LayerNorm over (C,H,W) per sample, in-place on z:  z = LN(z + y) * w + b
// ---------------------------------------------------------------------------
__global__ void ln_kernel(float* __restrict__ z, const float* __restrict__ y,
                          const float* __restrict__ w, const float* __restrict__ bb)
{
    const int b = blockIdx.x;
    const int tid = threadIdx.x;
    const int N = 64 * 256;
    float* zb = z + b * N;
    const float* yb = y + b * N;

    float s = 0.f, s2 = 0.f;
    for (int i = tid; i < N; i += 256) {
        float t = zb[i] + yb[i];
        s += t; s2 += t * t;
    }
    __shared__ float rs[256], rq[256];
    rs[tid] = s; rq[tid] = s2;
    __syncthreads();
    for (int o = 128; o > 0; o >>= 1) {
        if (tid < o) { rs[tid] += rs[tid + o]; rq[tid] += rq[tid + o]; }
        __syncthreads();
    }
    const float mean = rs[0] / N;
    const float var  = rq[0] / N - mean * mean;
    const float inv  = 1.0f / sqrtf(var + 1e-5f);
    for (int i = tid; i < N; i += 256) {
        float t = zb[i] + yb[i];
        zb[i] = (t - mean) * inv * w[i] + bb[i];
    }
}

// frame (B,1,64,64) -> d_out[:, j]  with output shape (B, pred_len, 1, 64, 64)
__global__ void copy_frame(const float* __restrict__ frame, float* __restrict__ out,
                           int pred_len, int j)
{
    int i = blockIdx.x * blockDim.x + threadIdx.x;   // 8192
    if (i >= 2 * 4096) return;
    int b = i >> 12, q = i & 4095;
    out[((long)(b * pred_len + j)) * 4096 + q] = frame[i];
}

// ---------------------------------------------------------------------------
// Host side
// ---------------------------------------------------------------------------
struct Res { const float *w1, *b1, *g1, *e1, *w2, *b2, *g2, *e2; };
struct Mam { const float *wd, *bd, *wB, *wC, *Dp, *logA, *dt; };

static void launch_conv(hipStream_t s, const float* x, long xbs,
    const float* w, const float* bias, const float* g, const float* e,
    const float* res, long rbs, const float* extra,
    float* out, long obs,
    int Cin, int Cout, int Hin, int Win, int Hout, int Wout,
    int KH, int KW, int stride, int pad, int flags)
{
    const int Bn = 2;
    const bool dualm = ((Cout & 31) == 0) && (Cin >= 4);
    const int Mtiles = dualm ? (Cout >> 5) : ((Cout + 15) >> 4);
    int tiles = Mtiles * ((Bn * Hout * Wout) / 16);
    const int wpb = 8;                       // 8 waves (256 threads) per block
    int blocks = (tiles + wpb - 1) / wpb;
    conv_wmma_kernel<<<blocks, wpb * 32, 0, s>>>(
        x, xbs, w, bias, g, e, res, rbs, extra, out, obs,
        Bn, Cin, Cout, Hin, Win, Hout, Wout, KH, KW, stride, pad, flags);
}

extern "C" void kernel_launch(void* const* d_in, const int* in_sizes, int n_in,
                              void* d_out, int out_size, void* d_ws, size_t ws_size,
                              hipStream_t stream)
{
    if (n_in < 83 || !d_out || !d_ws) return;

    // ---- unpack inputs (setup_inputs dict order, recursively flattened) ----
    int ki = 0;
    auto P = [&]() { return (const float*)d_in[ki++]; };
    const float* input_seq = P();                               // 0
    const float* enc_w1 = P(); const float* enc_b1 = P();
    const float* enc_g1 = P(); const float* enc_e1 = P();       // 1-4
    Res enc_res1 = {P(),P(),P(),P(),P(),P(),P(),P()};           // 5-12
    const float* enc_w2 = P(); const float* enc_b2 = P();
    const float* enc_g2 = P(); const float* enc_e2 = P();       // 13-16
    Res enc_res2 = {P(),P(),P(),P(),P(),P(),P(),P()};           // 17-24
    Res enc_res3 = {P(),P(),P(),P(),P(),P(),P(),P()};           // 25-32
    Mam mam[3];
    for (int i = 0; i < 3; ++i) mam[i] = {P(),P(),P(),P(),P(),P(),P()}; // 33-53
    const float *lnw[3], *lnb[3];
    for (int i = 0; i < 3; ++i) { lnw[i] = P(); lnb[i] = P(); } // 54-59
    Res dec_res1 = {P(),P(),P(),P(),P(),P(),P(),P()};           // 60-67
    const float* dec_tw1 = P(); const float* dec_tb1 = P();
    const float* dec_g1  = P(); const float* dec_e1  = P();     // 68-71
    Res dec_res2 = {P(),P(),P(),P(),P(),P(),P(),P()};           // 72-79
    const float* dec_tw2 = P(); const float* dec_tb2 = P();     // 80-81
    (void)in_sizes;                                             // 82: pred_len (derived below)

    // ---- workspace layout ----
    float* base = (float*)d_ws;
    size_t off = 0;
    auto alloc = [&](size_t n) { float* p = base + off; off += n; return p; };
    float* frame = alloc(2 * 4096);            // (B,1,64,64)
    float* e1    = alloc(2 * 32 * 1024);       // (B,32,32,32)
    float* e1t   = alloc(2 * 32 * 1024);
    float* z     = alloc(2 * 64 * 256);        // (B,64,16,16)
    float* zt    = alloc(2 * 64 * 256);
    float* ybuf  = alloc(2 * 64 * 256);
    float* dbuf  = alloc(2 * 64 * 256);        // delta
    float* Bv    = alloc(2 * 8 * 256);
    float* Cvb   = alloc(2 * 8 * 256);
    float* Abar  = alloc(2 * 64 * 8 * 256);
    float* Ebuf  = alloc(2 * 64 * 8 * 256);
    const size_t STATE = (size_t)2 * 49 * 64 * 8 * 256;   // 12,845,056 floats
    float* q[4];
    for (int i = 0; i < 4; ++i) q[i] = alloc(STATE);      // 3 live + 1 rotating
    if (off * sizeof(float) > ws_size) return;            // workspace too small

    float* cur[3] = { q[0], q[1], q[2] };
    float* freeb  = q[3];

    const long ZBS = 64 * 256;     // latent batch stride
    const long EBS = 32 * 1024;    // 32ch/32x32 batch stride

    auto resblock = [&](float* buf, float* tmp, const Res& r, int C, int H, int W) {
        long bs = (long)C * H * W;
        launch_conv(stream, buf, bs, r.w1, r.b1, r.g1, r.e1, nullptr, 0, nullptr,
                    tmp, bs, C, C, H, W, H, W, 3, 3, 1, 1, F_BN | F_LRELU);
        launch_conv(stream, tmp, bs, r.w2, r.b2, r.g2, r.e2, buf, bs, nullptr,
                    buf, bs, C, C, H, W, H, W, 3, 3, 1, 1, F_BN | F_ADD | F_LRELU);
    };

    auto encoder = [&](const float* xin, long xbs) {
        launch_conv(stream, xin, xbs, enc_w1, enc_b1, enc_g1, enc_e1, nullptr, 0, nullptr,
                    e1, EBS, 1, 32, 64, 64, 32, 32, 4, 4, 2, 1, F_BN | F_LRELU);
        resblock(e1, e1t, enc_res1, 32, 32, 32);
        launch_conv(stream, e1, EBS, enc_w2, enc_b2, enc_g2, enc_e2, nullptr, 0, nullptr,
                    z, ZBS, 32, 64, 32, 32, 16, 16, 4, 4, 2, 1, F_BN | F_LRELU);
        resblock(z, zt, enc_res2, 64, 16, 16);
        resblock(z, zt, enc_res3, 64, 16, 16);
    };

    auto stack = [&](bool first) {
        for (int i = 0; i < 3; ++i) {
            const Mam& m = mam[i];
            // delta = softplus(conv_circ(u,wd)+bd+dt)
            launch_conv(stream, z, ZBS, m.wd, m.bd, nullptr, nullptr, nullptr, 0, m.dt,
                        dbuf, ZBS, 64, 64, 16, 16, 16, 16, 3, 3, 1, 1,
                        F_CIRC | F_SOFTPLUS);
            launch_conv(stream, z, ZBS, m.wB, nullptr, nullptr, nullptr, nullptr, 0, nullptr,
                        Bv, 8 * 256L, 64, 8, 16, 16, 16, 16, 3, 3, 1, 1, F_CIRC);
            launch_conv(stream, z, ZBS, m.wC, nullptr, nullptr, nullptr, nullptr, 0, nullptr,
                        Cvb, 8 * 256L, 64, 8, 16, 16, 16, 16, 3, 3, 1, 1, F_CIRC);
            mamba_prep<<<1024, 256, 0, stream>>>(dbuf, Bv, z, m.logA, Abar, Ebuf);
            mamba_state<<<128, 256, 0, stream>>>(cur[i], freeb, Abar, Ebuf, Cvb,
                                                 z, m.Dp, ybuf, first ? 1 : 0);
            float* old = cur[i]; cur[i] = freeb; freeb = old;   // rotate buffers
            ln_kernel<<<2, 256, 0, stream>>>(z, ybuf, lnw[i], lnb[i]);
        }
    };

    auto decoder = [&]() {
        resblock(z, zt, dec_res1, 64, 16, 16);
        launch_conv(stream, z, ZBS, dec_tw1, dec_tb1, dec_g1, dec_e1, nullptr, 0, nullptr,
                    e1, EBS, 64, 32, 16, 16, 32, 32, 4, 4, 2, 1,
                    F_TRANS | F_BN | F_LRELU);
        resblock(e1, e1t, dec_res2, 32, 32, 32);
        launch_conv(stream, e1, EBS, dec_tw2, dec_tb2, nullptr, nullptr, nullptr, 0, nullptr,
                    frame, 4096L, 32, 1, 32, 32, 64, 64, 4, 4, 2, 1,
                    F_TRANS | F_SIGMOID);
    };

    // ---- sequence: 3 encode steps, then autoregressive prediction ----
    bool first = true;
    for (int t = 0; t < 3; ++t) {
        encoder(input_seq + (long)t * 4096, 3L * 4096);   // (B,T,1,64,64)
        stack(first);
        first = false;
    }
    decoder();                                            // last_x

    int pred_len = out_size / (2 * 4096);
    if (pred_len < 1) pred_len = 1;
    for (int j = 0; j < pred_len; ++j) {
        encoder(frame, 4096L);
        stack(false);
        decoder();
        copy_frame<<<32, 256, 0, stream>>>(frame, (float*)d_out, pred_len, j);
    }
}